// CustomGate_50259707298060
// MI455X (gfx1250) — compile-verified
//
#include <hip/hip_runtime.h>

// Qudit gate: y[a,i,r,b] = sum_j M[i,j] * x[a,j,r,b]  (complex, fp32)
// 64 independent complex GEMMs: Y_a (64 x 8192) = M (64x64) @ X_a (64 x 8192).
//
// Roofline: 17.2 GFLOP vs ~537 MB HBM traffic -> ~23us @ 23.3 TB/s, demanding
// ~750 TFLOP/s fp32 -> V_WMMA_F32_16X16X4_F32, single-pass LDS tiling (each x
// element hits HBM once, 4x reuse across i-tiles). X is staged through the
// ASYNCcnt path (GLOBAL_LOAD_ASYNC_TO_LDS_B32, no VGPR round-trip) into a
// K-interleaved layout so every B fragment is one aligned ds b64 that lands
// directly as a WMMA operand pair (no register shuffles, 2addr_b64-mergeable).

typedef __attribute__((ext_vector_type(2))) float v2f;
typedef __attribute__((ext_vector_type(8))) float v8f;

#define DIMQ   64
#define NCOLS  8192          // 64 (r) * 128 (batch) columns per a-slice
#define ASTRIDE (DIMQ * NCOLS)
#define BLK_N  128           // columns per block tile

__global__ __launch_bounds__(256)
void qudit_gate_wmma(const float* __restrict__ Mr,
                     const float* __restrict__ Mi,
                     const float* __restrict__ Xr,
                     const float* __restrict__ Xi,
                     float2* __restrict__ Y) {
    // LDS (64 KB): two 64x128 fp32 tiles, K-interleaved:
    //   element (j, n) lives at ((j/4)*128 + n)*4 + slot(j%4), slot = {0,2,1,3}
    // so floats [.., X[4jb+k0][n], X[4jb+k0+2][n], ..] are an aligned b64 pair
    // = exactly the B fragment (component v -> K = 2v + k0) for lane-half k0.
    extern __shared__ float smem[];
    float* sXr = smem;                   // 64*128 floats, interleaved
    float* sXi = smem + DIMQ * BLK_N;

    const int a    = blockIdx.y;                 // 0..63 (left wire index)
    const int nc0  = blockIdx.x * BLK_N;         // column tile base
    const int tid  = threadIdx.x;
    const int lane = tid & 31;
    const int wave = tid >> 5;

    // ---- stage X tile (re+im): async global->LDS b32, one cacheline/issue ----
    {
        const float* gr = Xr + (size_t)a * ASTRIDE + nc0;
        const float* gi = Xi + (size_t)a * ASTRIDE + nc0;
        const int n  = tid & 127;                // lane-contiguous columns
        const int jb0 = tid >> 7;                // 0 or 1
        #pragma unroll
        for (int iter = 0; iter < 32; ++iter) {
            const int j    = jb0 + iter * 2;     // row 0..63
            const int slot = ((j & 1) << 1) | ((j >> 1) & 1);
            const int loff = (((j >> 2) * BLK_N + n) << 2) + slot;
            const float* gpr = gr + (size_t)j * NCOLS + n;
            const float* gpi = gi + (size_t)j * NCOLS + n;
            // Flat->LDS truncates to addr[31:0]: low 32 bits of the generic
            // shared pointer serve as the DS destination address.
            asm volatile("global_load_async_to_lds_b32 %0, %1, off"
                         :: "v"((unsigned)(uintptr_t)(sXr + loff)), "v"(gpr)
                         : "memory");
            asm volatile("global_load_async_to_lds_b32 %0, %1, off"
                         :: "v"((unsigned)(uintptr_t)(sXi + loff)), "v"(gpi)
                         : "memory");
        }
    }

    // ---- per-wave tile assignment ----
    const int it  = wave & 3;   const int i0  = it << 4;   // i-tile (rows of M)
    const int nh  = wave >> 2;  const int nw0 = nh << 6;   // 64-col half
    const int m   = lane & 15;
    const int k0  = lane >> 4;                             // lane-half select

    // ---- A fragments (ISA 7.12.2, 32-bit A 16x4):
    //      lane L, component v -> A[M = L%16][K = 2*(L/16) + v]
    //      Hold Mr, Mi, -Mi: acc_r += Mr*xr + (-Mi)*xi ; acc_i += Mr*xi + Mi*xr
    v2f ar[16], ai[16], an[16];
    #pragma unroll
    for (int ks = 0; ks < 16; ++ks) {
        const int j0 = ks * 4;
        ar[ks] = *(const v2f*)(Mr + (i0 + m) * DIMQ + j0 + 2 * k0);
        ai[ks] = *(const v2f*)(Mi + (i0 + m) * DIMQ + j0 + 2 * k0);
        an[ks] = -ai[ks];
    }

    asm volatile("s_wait_asynccnt 0x0" ::: "memory");
    __syncthreads();

    v8f accr0 = {}, accr1 = {}, accr2 = {}, accr3 = {};
    v8f acci0 = {}, acci1 = {}, acci2 = {}, acci3 = {};

    const int coff = (nw0 + m) << 2;     // interleaved float offset of column

    // ---- K loop, fully unrolled: 16 K-blocks of K=4, 16 WMMAs per block ----
    #pragma unroll
    for (int ks = 0; ks < 16; ++ks) {
        const float* pr = sXr + ((ks * BLK_N) << 2) + (k0 << 1) + coff;
        const float* pi = sXi + ((ks * BLK_N) << 2) + (k0 << 1) + coff;
        const v2f A_r = ar[ks];
        const v2f A_i = ai[ks];
        const v2f A_n = an[ks];

        #define SUBTILE(S, ACR, ACI)                                              \
        {                                                                         \
            v2f br = *(const v2f*)(pr + (S) * 64);  /* aligned b64 B fragment */  \
            v2f bi = *(const v2f*)(pi + (S) * 64);                                \
            ACR = __builtin_amdgcn_wmma_f32_16x16x4_f32(false, A_r, false, br,    \
                                                        (short)0, ACR, false, false); \
            ACR = __builtin_amdgcn_wmma_f32_16x16x4_f32(false, A_n, false, bi,    \
                                                        (short)0, ACR, false, false); \
            ACI = __builtin_amdgcn_wmma_f32_16x16x4_f32(false, A_r, false, bi,    \
                                                        (short)0, ACI, false, false); \
            ACI = __builtin_amdgcn_wmma_f32_16x16x4_f32(false, A_i, false, br,    \
                                                        (short)0, ACI, false, false); \
        }
        SUBTILE(0, accr0, acci0)
        SUBTILE(1, accr1, acci1)
        SUBTILE(2, accr2, acci2)
        SUBTILE(3, accr3, acci3)
        #undef SUBTILE
    }

    // ---- store: C/D layout (VGPR v: M = v + 8*(L/16), N = L%16),
    //      interleaved complex64 output, coalesced b64 stores ----
    const size_t ybase = (size_t)a * ASTRIDE + (size_t)nc0 + nw0 + m;
    #pragma unroll
    for (int v = 0; v < 8; ++v) {
        const int i = i0 + v + 8 * k0;
        float2* yr = Y + ybase + (size_t)i * NCOLS;
        yr[0]  = make_float2(accr0[v], acci0[v]);
        yr[16] = make_float2(accr1[v], acci1[v]);
        yr[32] = make_float2(accr2[v], acci2[v]);
        yr[48] = make_float2(accr3[v], acci3[v]);
    }
}

extern "C" void kernel_launch(void* const* d_in, const int* in_sizes, int n_in,
                              void* d_out, int out_size, void* d_ws, size_t ws_size,
                              hipStream_t stream) {
    const float* Mr = (const float*)d_in[0];   // (64, 64)
    const float* Mi = (const float*)d_in[1];   // (64, 64)
    const float* Xr = (const float*)d_in[2];   // (262144, 128)
    const float* Xi = (const float*)d_in[3];   // (262144, 128)
    float2* Y = (float2*)d_out;                // (262144, 128) complex64 interleaved

    dim3 grid(NCOLS / BLK_N, DIMQ, 1);         // 64 n-tiles x 64 a-slices
    dim3 block(256, 1, 1);                     // 8 waves (wave32)
    const size_t shmem = (size_t)2 * DIMQ * BLK_N * sizeof(float);  // 64 KB

    qudit_gate_wmma<<<grid, block, shmem, stream>>>(Mr, Mi, Xr, Xi, Y);
}